// LightGraphConv_19524921327938
// MI455X (gfx1250) — compile-verified
//
#include <hip/hip_runtime.h>
#include <hip/hip_bf16.h>

#define D_FEAT   64
#define CHUNK    256   // edges staged per block iteration
#define THREADS  256   // 8 waves (wave32)

// ---------------------------------------------------------------------------
// CDNA5 async global->LDS copy (ASYNCcnt path) + waits, via inline asm.
// LDS byte offset is taken from the vdst VGPR; global address is SGPR base plus VGPR byte offset
// ---------------------------------------------------------------------------
__device__ __forceinline__ void async_load_b32_to_lds(uint32_t lds_off,
                                                      uint32_t byte_off,
                                                      const void* base) {
    asm volatile("global_load_async_to_lds_b32 %0, %1, %2"
                 :
                 : "v"(lds_off), "v"(byte_off), "s"(base)
                 : "memory");
}

__device__ __forceinline__ void wait_async_le3() {
    asm volatile("s_wait_asynccnt 0x3" ::: "memory");
}
__device__ __forceinline__ void wait_async_0() {
    asm volatile("s_wait_asynccnt 0x0" ::: "memory");
}

// Truncating a generic pointer derived from a __shared__ object yields the
// 32-bit LDS byte offset (shared aperture keeps the offset in the low dword).
__device__ __forceinline__ uint32_t lds_offset_of(const void* p) {
    return (uint32_t)(uintptr_t)p;
}

// ---------------------------------------------------------------------------
// Zero-init the output (harness poisons d_out; scatter-add needs zeros).
// ---------------------------------------------------------------------------
__global__ void lgc_zero_kernel(float4* __restrict__ out, int n4) {
    int i = blockIdx.x * blockDim.x + threadIdx.x;
    if (i < n4) out[i] = make_float4(0.f, 0.f, 0.f, 0.f);
}

// ---------------------------------------------------------------------------
// COO SpMM scatter: out[rows[e], :] += vals[e] * feats[cols[e], :]
// 16 lanes (one float4 each) cover one 64-float feature row -> 2 edges/wave.
// Edge stream (rows/cols/vals) is double-buffered through LDS with async
// global->LDS loads so the random feature gathers own the vector-load path.
// ---------------------------------------------------------------------------
__global__ void __launch_bounds__(THREADS)
lgc_spmm_kernel(const int* __restrict__ rows,
                const int* __restrict__ cols,
                const float* __restrict__ vals,
                const float* __restrict__ feats,
                float* __restrict__ out,
                int E, int numChunks) {
    __shared__ int   s_rows[2][CHUNK];
    __shared__ int   s_cols[2][CHUNK];
    __shared__ float s_vals[2][CHUNK];

    const int tid  = threadIdx.x;
    const int grp  = tid >> 4;   // 16 edge-processors per block
    const int lane = tid & 15;   // float4 segment within the feature row

    int c = blockIdx.x;
    if (c >= numChunks) return;

    // stage one chunk of the edge stream into LDS buffer `buf` (3 async
    // b32 loads per wave; out-of-range edges clamp the address, and the
    // processing loop re-checks e < E so stale data is never used)
    auto stage = [&](int buf, int chunkIdx) {
        int e  = chunkIdx * CHUNK + tid;
        int eg = (e < E) ? e : (E - 1);
        uint32_t off = (uint32_t)eg * 4u;
        async_load_b32_to_lds(lds_offset_of(&s_rows[buf][tid]), off, rows);
        async_load_b32_to_lds(lds_offset_of(&s_cols[buf][tid]), off, cols);
        async_load_b32_to_lds(lds_offset_of(&s_vals[buf][tid]), off, vals);
    };

    int buf = 0;
    stage(0, c);
    int next = c + gridDim.x;

    for (;;) {
        const bool haveNext = (next < numChunks);
        if (haveNext) {
            stage(buf ^ 1, next);   // overlap next-chunk staging with compute
            wait_async_le3();       // current chunk's 3 loads are complete
        } else {
            wait_async_0();
        }
        __syncthreads();

        const int base_e = c * CHUNK;
#pragma unroll 4
        for (int j = grp; j < CHUNK; j += 16) {
            const int e = base_e + j;
            if (e < E) {
                const int   r  = s_rows[buf][j];
                const int   cc = s_cols[buf][j];
                const float v  = s_vals[buf][j];

                // prefetch the feature row of this group's next edge
                if (j + 16 < CHUNK && e + 16 < E) {
                    const int cn = s_cols[buf][j + 16];
                    __builtin_prefetch(feats + (size_t)cn * D_FEAT + lane * 4, 0, 0);
                }

                // gather 16 B of the 256 B feature row (global_load_b128)
                const float4 f =
                    *(reinterpret_cast<const float4*>(feats + (size_t)cc * D_FEAT) + lane);

                float* dst = out + (size_t)r * D_FEAT + lane * 4;
                unsafeAtomicAdd(dst + 0, v * f.x);
                unsafeAtomicAdd(dst + 1, v * f.y);
                unsafeAtomicAdd(dst + 2, v * f.z);
                unsafeAtomicAdd(dst + 3, v * f.w);
            }
        }
        __syncthreads();   // buffer `buf` free for reuse next iteration

        if (!haveNext) break;
        buf ^= 1;
        c = next;
        next += gridDim.x;
    }
}

// ---------------------------------------------------------------------------
// Launch
// ---------------------------------------------------------------------------
extern "C" void kernel_launch(void* const* d_in, const int* in_sizes, int n_in,
                              void* d_out, int out_size, void* d_ws, size_t ws_size,
                              hipStream_t stream) {
    const int*   adj_idx = (const int*)d_in[0];   // [2, E] int32
    const float* vals    = (const float*)d_in[1]; // [E]
    const float* feats   = (const float*)d_in[2]; // [N, 64]

    const int E = in_sizes[1];
    const int* rows = adj_idx;
    const int* cols = adj_idx + E;

    // 1) zero the output
    const int n4 = out_size / 4;
    lgc_zero_kernel<<<(n4 + THREADS - 1) / THREADS, THREADS, 0, stream>>>(
        (float4*)d_out, n4);

    // 2) scatter SpMM
    const int numChunks = (E + CHUNK - 1) / CHUNK;
    int grid = numChunks < 1536 ? numChunks : 1536;  // >=2 chunks/block for pipelining
    lgc_spmm_kernel<<<grid, THREADS, 0, stream>>>(rows, cols, vals, feats,
                                                  (float*)d_out, E, numChunks);
}